// Fast_Tucker_Taylor_26113401160148
// MI455X (gfx1250) — compile-verified
//
#include <hip/hip_runtime.h>

typedef __attribute__((ext_vector_type(2))) float v2f;
typedef __attribute__((ext_vector_type(8))) float v8f;

#define BATCH 4096
#define INF   1024
#define OUTF  1024

__device__ __forceinline__ v8f wmma_f32(v2f a, v2f b, v8f c) {
  // V_WMMA_F32_16X16X4_F32 : D = A(16x4) * B(4x16) + C(16x16), all fp32
  return __builtin_amdgcn_wmma_f32_16x16x4_f32(false, a, false, b, (short)0, c,
                                               false, false);
}

// ---------------------------------------------------------------------------
// Stage 1: Z(192 x 4096) = [I0; I1; I2] (192 x 1024) @ X^T (1024 x 4096)
// 12 x 256 = 3072 tiles, 8 waves/block -> 384 blocks (exact, no EXEC holes)
// ---------------------------------------------------------------------------
__global__ void __launch_bounds__(256) taylor_z_kernel(
    const float* __restrict__ X, const float* __restrict__ I0,
    const float* __restrict__ I1, const float* __restrict__ I2,
    float* __restrict__ Z) {
  const int lane = threadIdx.x & 31;
  const int wave = threadIdx.x >> 5;
  const int tile = blockIdx.x * 8 + wave;
  const int mt = tile % 12, nt = tile / 12;
  const int row0 = mt * 16, col0 = nt * 16;

  const float* Arow;
  if (row0 < 32)      Arow = I0 + row0 * INF;
  else if (row0 < 96) Arow = I1 + (row0 - 32) * INF;
  else                Arow = I2 + (row0 - 96) * INF;

  const int m = lane & 15, n = lane & 15;
  const int ko = (lane >> 4) * 2;
  const float* ap = Arow + m * INF + ko;
  const float* bp = X + (col0 + n) * INF + ko;  // B[k][n] = X[n][k]

  v8f acc = {};
  for (int k = 0; k < INF; k += 4) {
    v2f a, b;
    a[0] = ap[k]; a[1] = ap[k + 1];
    b[0] = bp[k]; b[1] = bp[k + 1];
    acc = wmma_f32(a, b, acc);
  }
  const int mb = (lane >> 4) * 8;
#pragma unroll
  for (int v = 0; v < 8; ++v)
    Z[(row0 + mb + v) * BATCH + col0 + n] = acc[v];
}

// ---------------------------------------------------------------------------
// Stage 2a: R0(32 x 4096) = G0(32x32) @ Z[0:32]   (K = 32)
// ---------------------------------------------------------------------------
__global__ void __launch_bounds__(256) taylor_r0_kernel(
    const float* __restrict__ G0, const float* __restrict__ Z,
    float* __restrict__ R) {
  const int lane = threadIdx.x & 31;
  const int wave = threadIdx.x >> 5;
  const int tile = blockIdx.x * 8 + wave;     // 2 x 256 = 512 tiles
  const int mt = tile & 1, nt = tile >> 1;
  const int row0 = mt * 16, col0 = nt * 16;
  const int m = lane & 15, n = lane & 15;
  const int ko = (lane >> 4) * 2;

  v8f acc = {};
  for (int k = 0; k < 32; k += 4) {
    const int kk = k + ko;
    v2f a, b;
    a[0] = G0[(row0 + m) * 32 + kk];
    a[1] = G0[(row0 + m) * 32 + kk + 1];
    b[0] = Z[kk * BATCH + col0 + n];
    b[1] = Z[(kk + 1) * BATCH + col0 + n];
    acc = wmma_f32(a, b, acc);
  }
  const int mb = (lane >> 4) * 8;
#pragma unroll
  for (int v = 0; v < 8; ++v)
    R[(row0 + mb + v) * BATCH + col0 + n] = acc[v];
}

// ---------------------------------------------------------------------------
// Stage 2b: R1 = G1(32x1024) @ kron1, kron1[k,b] = Z[64+k/32][b]*Z[32+k%32][b]
// kron fused into B-fragment construction (never materialized).
// ---------------------------------------------------------------------------
__global__ void __launch_bounds__(256) taylor_r1_kernel(
    const float* __restrict__ G1, const float* __restrict__ Z,
    float* __restrict__ R) {
  const int lane = threadIdx.x & 31;
  const int wave = threadIdx.x >> 5;
  const int tile = blockIdx.x * 8 + wave;     // 2 x 256 = 512 tiles
  const int mt = tile & 1, nt = tile >> 1;
  const int row0 = mt * 16, col0 = nt * 16;
  const int m = lane & 15, n = lane & 15;
  const int ko = (lane >> 4) * 2;
  const int c = col0 + n;

  v8f acc = {};
  for (int k = 0; k < 1024; k += 4) {
    const int k0 = k + ko, k1 = k0 + 1;
    v2f a, b;
    a[0] = G1[(row0 + m) * 1024 + k0];
    a[1] = G1[(row0 + m) * 1024 + k1];
    b[0] = Z[(64 + (k0 >> 5)) * BATCH + c] * Z[(32 + (k0 & 31)) * BATCH + c];
    b[1] = Z[(64 + (k1 >> 5)) * BATCH + c] * Z[(32 + (k1 & 31)) * BATCH + c];
    acc = wmma_f32(a, b, acc);
  }
  const int mb = (lane >> 4) * 8;
#pragma unroll
  for (int v = 0; v < 8; ++v)
    R[(32 + row0 + mb + v) * BATCH + col0 + n] = acc[v];
}

// ---------------------------------------------------------------------------
// Stage 2c: R2 = G2(32x32768) @ kron2 (K = 32768, 8.6 GFLOP -> dominant GEMM)
// Block = 1 column tile: Z factor rows 96..191 staged in LDS (6 KB), shared
// by 8 waves = 2 row tiles x 4 split-K slices; partials combined with
// global_atomic_add_f32 into pre-zeroed R.
// ---------------------------------------------------------------------------
__global__ void __launch_bounds__(256) taylor_r2_kernel(
    const float* __restrict__ G2, const float* __restrict__ Z,
    float* __restrict__ R) {
  __shared__ float zs[3 * 32 * 16];
  const int lane = threadIdx.x & 31;
  const int wave = threadIdx.x >> 5;
  const int col0 = blockIdx.x * 16;

  for (int idx = threadIdx.x; idx < 1536; idx += 256) {
    const int rr = idx >> 4, cc = idx & 15;
    zs[idx] = Z[(96 + rr) * BATCH + col0 + cc];
  }
  __syncthreads();

  const int rt = wave >> 2;          // row tile 0..1
  const int ks = wave & 3;           // split-K slice 0..3
  const int row0 = rt * 16;
  const int m = lane & 15, n = lane & 15;
  const int ko = (lane >> 4) * 2;
  const int kbeg = ks * 8192, kend = kbeg + 8192;
  const float* gp = G2 + (row0 + m) * 32768;

  v8f acc = {};
  for (int k = kbeg; k < kend; k += 4) {
    const int k0 = k + ko, k1 = k0 + 1;
    v2f a, b;
    a[0] = gp[k0];
    a[1] = gp[k1];
    b[0] = zs[(k0 & 31) * 16 + n] *
           zs[512 + ((k0 >> 5) & 31) * 16 + n] *
           zs[1024 + (k0 >> 10) * 16 + n];
    b[1] = zs[(k1 & 31) * 16 + n] *
           zs[512 + ((k1 >> 5) & 31) * 16 + n] *
           zs[1024 + (k1 >> 10) * 16 + n];
    acc = wmma_f32(a, b, acc);
  }
  const int mb = (lane >> 4) * 8;
#pragma unroll
  for (int v = 0; v < 8; ++v)
    atomicAdd(&R[(64 + row0 + mb + v) * BATCH + col0 + n], acc[v]);
}

// ---------------------------------------------------------------------------
// Stage 3: Y(4096 x 1024) = (Ocat(1024x96) @ R(96x4096))^T + const
// 64 x 256 = 16384 tiles -> 2048 blocks; padded-LDS transpose for coalesced
// (batch-major) stores.
// ---------------------------------------------------------------------------
__global__ void __launch_bounds__(256) taylor_out_kernel(
    const float* __restrict__ O0, const float* __restrict__ O1,
    const float* __restrict__ O2, const float* __restrict__ Cst,
    const float* __restrict__ R, float* __restrict__ Y) {
  __shared__ float ts[8 * 16 * 17];
  const int lane = threadIdx.x & 31;
  const int wave = threadIdx.x >> 5;
  const int tile = blockIdx.x * 8 + wave;
  const int mt = tile & 63, nt = tile >> 6;
  const int row0 = mt * 16;          // out-feature
  const int col0 = nt * 16;          // batch
  const int m = lane & 15, n = lane & 15;
  const int ko = (lane >> 4) * 2;

  v8f acc = {};
  for (int k = 0; k < 96; k += 4) {
    const int kk = k + ko;           // kk even; kk,kk+1 stay in one segment
    const float* Op = O0; int rk = kk;
    if (kk >= 64)      { Op = O2; rk = kk - 64; }
    else if (kk >= 32) { Op = O1; rk = kk - 32; }
    v2f a, b;
    a[0] = Op[(row0 + m) * 32 + rk];
    a[1] = Op[(row0 + m) * 32 + rk + 1];
    b[0] = R[kk * BATCH + col0 + n];
    b[1] = R[(kk + 1) * BATCH + col0 + n];
    acc = wmma_f32(a, b, acc);
  }
  const int mb = (lane >> 4) * 8;
  float* tw = ts + wave * 272;
#pragma unroll
  for (int v = 0; v < 8; ++v)
    tw[(mb + v) * 17 + n] = acc[v];
  __syncthreads();

  const int ol = lane & 15;          // out-feature within tile
  const int bh = (lane >> 4) * 8;    // batch half within tile
  const float cv = Cst[row0 + ol];
#pragma unroll
  for (int v = 0; v < 8; ++v)
    Y[(col0 + bh + v) * OUTF + row0 + ol] = tw[ol * 17 + bh + v] + cv;
}

extern "C" void kernel_launch(void* const* d_in, const int* in_sizes, int n_in,
                              void* d_out, int out_size, void* d_ws,
                              size_t ws_size, hipStream_t stream) {
  const float* X  = (const float*)d_in[0];
  const float* C  = (const float*)d_in[1];
  const float* O0 = (const float*)d_in[2];
  const float* I0 = (const float*)d_in[3];
  const float* G0 = (const float*)d_in[4];
  const float* O1 = (const float*)d_in[5];
  const float* I1 = (const float*)d_in[6];
  const float* G1 = (const float*)d_in[7];
  const float* O2 = (const float*)d_in[8];
  const float* I2 = (const float*)d_in[9];
  const float* G2 = (const float*)d_in[10];
  float* Y = (float*)d_out;

  float* Z = (float*)d_ws;           // 192 x 4096 = 3 MB
  float* R = Z + 192 * BATCH;        // 96 x 4096  = 1.5 MB

  // Zero only the split-K accumulated region (R2 rows 64..95).
  hipMemsetAsync(R + 64 * BATCH, 0, 32 * BATCH * sizeof(float), stream);

  taylor_z_kernel  <<<384,  256, 0, stream>>>(X, I0, I1, I2, Z);
  taylor_r0_kernel <<<64,   256, 0, stream>>>(G0, Z, R);
  taylor_r1_kernel <<<64,   256, 0, stream>>>(G1, Z, R);
  taylor_r2_kernel <<<256,  256, 0, stream>>>(G2, Z, R);
  taylor_out_kernel<<<2048, 256, 0, stream>>>(O0, O1, O2, C, R, Y);
}